// RewardModel_73976516706801
// MI455X (gfx1250) — compile-verified
//
#include <hip/hip_runtime.h>
#include <hip/hip_bf16.h>
#include <math.h>

typedef __attribute__((ext_vector_type(2))) float v2f;
typedef __attribute__((ext_vector_type(8))) float v8f;

// ---------------------------------------------------------------------------
// Kernel 1: rewards[b*S+s] = dot(hidden[b,s,:], w)  via V_WMMA_F32_16X16X4_F32
// One wave32 per 16-row tile. B-matrix (4x16) = 16 hidden rows (N=lane&15,
// K split across lane halves per the 32-bit operand layout). A-matrix (16x4)
// = w chunk broadcast across all M rows -> every D row holds the 16 dots.
// Pure streaming: 268 MB read once, HBM-bound (~11.5 us floor at 23.3 TB/s).
// ---------------------------------------------------------------------------
__global__ __launch_bounds__(256) void rm_gemv_wmma(
    const float* __restrict__ hs, const float* __restrict__ w,
    float* __restrict__ rewards, int H, int tiles)
{
  const int lane = threadIdx.x & 31;
  const int wave = threadIdx.x >> 5;
  const int tile = blockIdx.x * 8 + wave;
  if (tile >= tiles) return;                 // wave-uniform: EXEC stays all-1s
  const int n    = lane & 15;                // column (row-of-hidden) index
  const int half = lane >> 4;                // K-half select
  const float* rowPtr = hs + ((size_t)tile * 16 + n) * (size_t)H;

  v8f c0 = {}; v8f c1 = {};                  // two accumulators break RAW chain
  for (int k0 = 0; k0 < H; k0 += 32) {
    // stream-ahead hint -> global_prefetch_b8 (speculative, safe at tensor end)
    __builtin_prefetch(rowPtr + k0 + 512, 0, 0);
#pragma unroll
    for (int j = 0; j < 8; ++j) {
      const int kk = k0 + 4 * j;
      // B fragment: lanes 0-15 get (K=kk,kk+1), lanes 16-31 get (K=kk+2,kk+3)
      v2f b = *(const v2f*)(rowPtr + kk + 2 * half);
      // A fragment: wave-uniform w chunk (compiler scalarizes to s_load)
      const float w0 = w[kk + 0], w1 = w[kk + 1], w2 = w[kk + 2], w3 = w[kk + 3];
      v2f a; a.x = half ? w2 : w0; a.y = half ? w3 : w1;
      if (j & 1)
        c1 = __builtin_amdgcn_wmma_f32_16x16x4_f32(false, a, false, b,
                                                   (short)0, c1, false, false);
      else
        c0 = __builtin_amdgcn_wmma_f32_16x16x4_f32(false, a, false, b,
                                                   (short)0, c0, false, false);
    }
  }
  // D[m=0][n] lives in accumulator VGPR0, lanes 0-15 (N = lane)
  const float r = c0[0] + c1[0];
  if (lane < 16) rewards[(size_t)tile * 16 + n] = r;
}

// ---------------------------------------------------------------------------
// Kernel 2: one block per pair. First-pad / first-divergence scans (min-tree),
// stable log-sigmoid sum over the selected range, chosen/rejected scores.
// ---------------------------------------------------------------------------
__global__ __launch_bounds__(256) void rm_pair(
    const int* __restrict__ ids, const float* __restrict__ rewards,
    const int* __restrict__ pad_ptr, float* __restrict__ out,
    float* __restrict__ per_sample, int S, int bs)
{
  const int p   = blockIdx.x;
  const int tid = threadIdx.x;
  const int pad = pad_ptr[0];
  const int*   cid = ids     + (size_t)p        * S;
  const int*   rid = ids     + (size_t)(bs + p) * S;
  const float* cr  = rewards + (size_t)p        * S;
  const float* rr  = rewards + (size_t)(bs + p) * S;

  __shared__ int sh_min[3][256];
  int cmin = S, rmin = S, dmin = S;
  for (int s = tid; s < S; s += 256) {
    const int cv = cid[s], rv = rid[s];
    if (cv == pad) cmin = min(cmin, s);
    if (rv == pad) rmin = min(rmin, s);
    if (cv != rv)  dmin = min(dmin, s);
  }
  sh_min[0][tid] = cmin; sh_min[1][tid] = rmin; sh_min[2][tid] = dmin;
  __syncthreads();
  for (int off = 128; off > 0; off >>= 1) {
    if (tid < off) {
      sh_min[0][tid] = min(sh_min[0][tid], sh_min[0][tid + off]);
      sh_min[1][tid] = min(sh_min[1][tid], sh_min[1][tid + off]);
      sh_min[2][tid] = min(sh_min[2][tid], sh_min[2][tid + off]);
    }
    __syncthreads();
  }
  const int  c_ind      = sh_min[0][0];
  const int  r_pad      = sh_min[1][0];
  const int  dfirst     = sh_min[2][0];
  const bool has_div    = dfirst < S;
  const int  divergence = has_div ? dfirst : (S - 1);
  const int  r_ind      = has_div ? r_pad : c_ind;
  const int  end_ind    = has_div ? max(c_ind, r_pad) : S;

  __shared__ float sh_sum[256];
  float lsum = 0.f;
  for (int s = divergence + tid; s < end_ind; s += 256) {
    const float x  = cr[s] - rr[s];
    // log_sigmoid(x) = min(x,0) - log1p(exp(-|x|))  (numerically stable)
    const float ls = fminf(x, 0.f) - log1pf(expf(-fabsf(x)));
    lsum += ls;
  }
  sh_sum[tid] = lsum;
  __syncthreads();
  for (int off = 128; off > 0; off >>= 1) {
    if (tid < off) sh_sum[tid] += sh_sum[tid + off];
    __syncthreads();
  }
  if (tid == 0) {
    int cnt = end_ind - divergence; if (cnt < 1) cnt = 1;
    per_sample[p] = -sh_sum[0] / (float)cnt;
    int ci = c_ind - 1; if (ci < 0) ci += S;   // JAX-style negative wrap
    int ri = r_ind - 1; if (ri < 0) ri += S;
    out[1 + p]      = cr[ci];
    out[1 + bs + p] = rr[ri];
  }
}

// ---------------------------------------------------------------------------
// Kernel 3: loss = mean(per_sample)
// ---------------------------------------------------------------------------
__global__ void rm_final(const float* __restrict__ ps, float* __restrict__ out,
                         int bs)
{
  if (threadIdx.x == 0 && blockIdx.x == 0) {
    float s = 0.f;
    for (int i = 0; i < bs; ++i) s += ps[i];
    out[0] = s / (float)bs;
  }
}

extern "C" void kernel_launch(void* const* d_in, const int* in_sizes, int n_in,
                              void* d_out, int out_size, void* d_ws, size_t ws_size,
                              hipStream_t stream) {
  const float* hs  = (const float*)d_in[0];   // (B,S,H) f32
  const int*   ids = (const int*)  d_in[1];   // (B,S)   i32
  const float* w   = (const float*)d_in[2];   // (H,)    f32
  const int*   pad = (const int*)  d_in[3];   // scalar

  float* out = (float*)d_out;

  const int H    = in_sizes[2];               // 4096
  const int rows = in_sizes[1];               // B*S = 16384
  const int bs   = (out_size - 1) / 2;        // 4
  const int B    = 2 * bs;                    // 8
  const int S    = rows / B;                  // 2048

  float* rewards    = (float*)d_ws;           // rows floats
  float* per_sample = rewards + rows;         // bs floats

  const int tiles  = rows / 16;               // 1024 wave-tiles
  const int blocks = (tiles + 7) / 8;         // 8 waves (256 thr) per block

  rm_gemv_wmma<<<blocks, 256, 0, stream>>>(hs, w, rewards, H, tiles);
  rm_pair<<<bs, 256, 0, stream>>>(ids, rewards, pad, out, per_sample, S, bs);
  rm_final<<<1, 32, 0, stream>>>(per_sample, out, bs);
}